// GNN_73581379715703
// MI455X (gfx1250) — compile-verified
//
#include <hip/hip_runtime.h>
#include <hip/hip_bf16.h>
#include <math.h>

typedef __attribute__((ext_vector_type(2))) float v2f;
typedef __attribute__((ext_vector_type(8))) float v8f;

#define N_NODES 50000
#define F_IN 128
#define F_HID 16
#define N_TILES (N_NODES / 16)   // 3125, exact

// ---------------------------------------------------------------------------
// Degree / normalization
// ---------------------------------------------------------------------------
__global__ void deg_init_kernel(float* __restrict__ deg, int n) {
    int i = blockIdx.x * blockDim.x + threadIdx.x;
    if (i < n) deg[i] = 1.0f;   // self-loop contribution
}

__global__ void deg_count_kernel(const long long* __restrict__ col,
                                 float* __restrict__ deg, int nEdges) {
    int i = blockIdx.x * blockDim.x + threadIdx.x;
    if (i < nEdges) unsafeAtomicAdd(&deg[(int)col[i]], 1.0f);
}

__global__ void deg_rsqrt_kernel(float* __restrict__ deg, int n) {
    int i = blockIdx.x * blockDim.x + threadIdx.x;
    if (i < n) deg[i] = rsqrtf(deg[i]);   // deg >= 1 always (self-loop)
}

// ---------------------------------------------------------------------------
// GEMM1: h1[N,16] = x[N,128] @ W1[128,16]  via V_WMMA_F32_16X16X4_F32
// One wave per 16-row tile; 32 K-steps of 4.
// ---------------------------------------------------------------------------
__global__ void gemm1_wmma_kernel(const float* __restrict__ x,
                                  const float* __restrict__ W1,
                                  float* __restrict__ h1) {
    int tile = blockIdx.x;                 // 0..3124
    int lane = threadIdx.x & 31;
    int half = lane >> 4;                  // 0: K pair {0,1}, 1: K pair {2,3}
    int lm   = lane & 15;                  // M (for A) / N (for B)
    int row  = tile * 16 + lm;

    const float* xr = x + row * F_IN;
    v8f c = {};
#pragma unroll 4
    for (int k = 0; k < 32; ++k) {
        int kb = k * 4 + 2 * half;
        v2f a = *(const v2f*)(xr + kb);              // A: M=lm, K={kb,kb+1}
        v2f b;
        b.x = W1[(kb + 0) * F_HID + lm];             // B: K rows, N=lm
        b.y = W1[(kb + 1) * F_HID + lm];
        c = __builtin_amdgcn_wmma_f32_16x16x4_f32(
                false, a, false, b, (short)0, c, false, false);
    }
#pragma unroll
    for (int v = 0; v < 8; ++v) {
        int m = v + 8 * half;                        // C/D layout
        h1[(tile * 16 + m) * F_HID + lm] = c[v];
    }
}

// ---------------------------------------------------------------------------
// Self-loop init: accum[i][f] = dinv[i]^2 * hin[i][f]
// ---------------------------------------------------------------------------
__global__ void self_init_kernel(const float* __restrict__ hin,
                                 const float* __restrict__ dinv,
                                 float* __restrict__ accum, int n) {
    int tid = blockIdx.x * blockDim.x + threadIdx.x;
    int i = tid >> 4, f = tid & 15;
    if (i < n) {
        float d = dinv[i];
        accum[i * F_HID + f] = d * d * hin[i * F_HID + f];
    }
}

// ---------------------------------------------------------------------------
// Edge aggregation: accum[col[e]][:] += dinv[row]*dinv[col] * hin[row[e]][:]
// 16 consecutive lanes share one edge -> gather & atomic each hit one 64B line.
// ---------------------------------------------------------------------------
__global__ void edge_agg_kernel(const long long* __restrict__ rowIdx,
                                const long long* __restrict__ colIdx,
                                const float* __restrict__ dinv,
                                const float* __restrict__ hin,
                                float* __restrict__ accum, int nEdges) {
    int tid = blockIdx.x * blockDim.x + threadIdx.x;
    int e = tid >> 4;
    if (e >= nEdges) return;
    int f = tid & 15;
    int r = (int)rowIdx[e];
    int c = (int)colIdx[e];
    float w = dinv[r] * dinv[c];
    unsafeAtomicAdd(&accum[c * F_HID + f], hin[r * F_HID + f] * w);
}

// ---------------------------------------------------------------------------
// Fused: h2 = ReLU(accum1 + b1);  accum2_init = dinv^2 * h2
// ---------------------------------------------------------------------------
__global__ void bias_relu_selfinit_kernel(const float* __restrict__ accum1,
                                          const float* __restrict__ b1,
                                          const float* __restrict__ dinv,
                                          float* __restrict__ h2,
                                          float* __restrict__ accum2, int n) {
    int tid = blockIdx.x * blockDim.x + threadIdx.x;
    int i = tid >> 4, f = tid & 15;
    if (i >= n) return;
    float v = fmaxf(accum1[i * F_HID + f] + b1[f], 0.0f);
    h2[i * F_HID + f] = v;
    float d = dinv[i];
    accum2[i * F_HID + f] = d * d * v;
}

// ---------------------------------------------------------------------------
// GEMM2 + bias + log_softmax:
// out[16rows,128] = log_softmax(agg2[16,16] @ W2[16,128] + b2)
// 256 threads = 8 waves; wave w computes the 16x16 tile at cols w*16..w*16+15
// (4 WMMA f32 16x16x4 steps), result staged in LDS, then width-16 shuffle
// reductions for the row-wise max / sum(exp).
// ---------------------------------------------------------------------------
__global__ void gemm2_softmax_kernel(const float* __restrict__ agg2,
                                     const float* __restrict__ W2,
                                     const float* __restrict__ b2,
                                     float* __restrict__ out) {
    __shared__ float tileLds[16 * F_IN];   // 8 KB

    int tileRow = blockIdx.x;              // 0..3124
    int t    = threadIdx.x;
    int wave = t >> 5;                     // 0..7 -> 16-col block
    int lane = t & 31;
    int half = lane >> 4;
    int lm   = lane & 15;
    int row  = tileRow * 16 + lm;

    v8f c = {};
#pragma unroll
    for (int k = 0; k < 4; ++k) {
        int kb = k * 4 + 2 * half;
        v2f a = *(const v2f*)(agg2 + row * F_HID + kb);
        v2f b;
        b.x = W2[(kb + 0) * F_IN + wave * 16 + lm];
        b.y = W2[(kb + 1) * F_IN + wave * 16 + lm];
        c = __builtin_amdgcn_wmma_f32_16x16x4_f32(
                false, a, false, b, (short)0, c, false, false);
    }
    float bias = b2[wave * 16 + lm];
#pragma unroll
    for (int v = 0; v < 8; ++v) {
        int m = v + 8 * half;
        tileLds[m * F_IN + wave * 16 + lm] = c[v] + bias;
    }
    __syncthreads();

    // log-softmax over 128 cols: 16 threads per row, 8 cols each
    int r   = t >> 4;                      // 0..15
    int seg = t & 15;                      // 0..15
    const float* rowp = tileLds + r * F_IN + seg * 8;

    float mx = -1e30f;
#pragma unroll
    for (int i = 0; i < 8; ++i) mx = fmaxf(mx, rowp[i]);
    for (int off = 8; off; off >>= 1) mx = fmaxf(mx, __shfl_xor(mx, off, 16));

    float s = 0.0f;
#pragma unroll
    for (int i = 0; i < 8; ++i) s += __expf(rowp[i] - mx);
    for (int off = 8; off; off >>= 1) s += __shfl_xor(s, off, 16);

    float lz = mx + __logf(s);
    float* op = out + (tileRow * 16 + r) * F_IN + seg * 8;
#pragma unroll
    for (int i = 0; i < 8; ++i) op[i] = rowp[i] - lz;
}

// ---------------------------------------------------------------------------
extern "C" void kernel_launch(void* const* d_in, const int* in_sizes, int n_in,
                              void* d_out, int out_size, void* d_ws, size_t ws_size,
                              hipStream_t stream) {
    const float*     x   = (const float*)d_in[0];
    const long long* ei  = (const long long*)d_in[1];   // int64 [2, E]
    const float*     W1  = (const float*)d_in[2];
    const float*     b1  = (const float*)d_in[3];
    const float*     W2  = (const float*)d_in[4];
    const float*     b2  = (const float*)d_in[5];
    float*           out = (float*)d_out;

    const int E = in_sizes[1] / 2;                       // 600000
    const long long* rowIdx = ei;                        // sources
    const long long* colIdx = ei + E;                    // targets

    // workspace carve-up (floats); all offsets 16B-aligned
    float* ws    = (float*)d_ws;
    float* dinv  = ws;                                   // 50000  (deg -> dinv in place)
    float* h1    = ws + 65536;                           // 800000
    float* agg1  = ws + 65536 + 800000;                  // 800000
    float* h2    = ws + 65536 + 1600000;                 // 800000
    float* agg2  = ws + 65536 + 2400000;                 // 800000
    (void)ws_size; (void)n_in; (void)out_size;

    const int nf = N_NODES * F_HID;                      // 800000

    // 1) degrees (self-loop = 1) and D^-1/2
    deg_init_kernel<<<(N_NODES + 255) / 256, 256, 0, stream>>>(dinv, N_NODES);
    deg_count_kernel<<<(E + 255) / 256, 256, 0, stream>>>(colIdx, dinv, E);
    deg_rsqrt_kernel<<<(N_NODES + 255) / 256, 256, 0, stream>>>(dinv, N_NODES);

    // 2) h1 = x @ W1   (WMMA f32)
    gemm1_wmma_kernel<<<N_TILES, 32, 0, stream>>>(x, W1, h1);

    // 3) agg1 = S @ h1  (self-loops folded into init, edges via f32 atomics)
    self_init_kernel<<<(nf + 255) / 256, 256, 0, stream>>>(h1, dinv, agg1, N_NODES);
    edge_agg_kernel<<<(E * 16 + 255) / 256, 256, 0, stream>>>(rowIdx, colIdx, dinv,
                                                              h1, agg1, E);

    // 4) h2 = ReLU(agg1 + b1); seed agg2 with self-loops
    bias_relu_selfinit_kernel<<<(nf + 255) / 256, 256, 0, stream>>>(agg1, b1, dinv,
                                                                    h2, agg2, N_NODES);

    // 5) agg2 = S @ h2   (aggregate at width 16, NOT 128: S(h2 W2) == (S h2) W2)
    edge_agg_kernel<<<(E * 16 + 255) / 256, 256, 0, stream>>>(rowIdx, colIdx, dinv,
                                                              h2, agg2, E);

    // 6) out = log_softmax(agg2 @ W2 + b2)   (WMMA f32 + fused softmax)
    gemm2_softmax_kernel<<<N_TILES, 256, 0, stream>>>(agg2, W2, b2, out);
}